// AdaptiveWeightedKnowledgeDistillationLoss_5703716569717
// MI455X (gfx1250) — compile-verified
//
#include <hip/hip_runtime.h>
#include <cstdint>

// ---------------------------------------------------------------------------
// AdaptiveWeightedKnowledgeDistillationLoss for MI455X (gfx1250)
//
// Roofline: 2 x [4096 x 32000] fp32 = 1.05 GB @ 23.3 TB/s => ~45 us floor.
// No GEMM structure -> WMMA inapplicable; the MI455X win is the async-tensor
// data path: per-wave global->LDS async copies (ASYNCcnt) double-buffered in
// 16 KB chunks, overlapped with a single-pass ONLINE softmax accumulation
// (running max + rescale), so HBM is read exactly once and load/compute fully
// overlap. Each lane reads only LDS it loaded itself -> the hot loop has NO
// barriers, only per-wave s_wait_asynccnt partial waits (in-order completion).
// ---------------------------------------------------------------------------

#define KD_T       4.0f
#define KD_INVT    0.25f
#define KD_C       32000
#define KD_NV4     (KD_C / 4)        // 8000 float4 per row
#define KD_CHUNKV4 1000              // float4 per chunk  (4000 cols, 16 KB/tensor)
#define KD_NCHUNK  8                 // 8 * 1000 = 8000 float4
#define KD_BDIM    512               // 16 wave32s per block
#define KD_NEGBIG  (-1.0e30f)        // finite sentinel: avoids inf-inf NaN in merges

// Builtin signature (probe-confirmed): pointee type is a 16B int vector.
typedef int v4i __attribute__((ext_vector_type(4)));
typedef __attribute__((address_space(1))) v4i* g_v4i_ptr;   // global
typedef __attribute__((address_space(3))) v4i* l_v4i_ptr;   // LDS

// ---- CDNA5 async global->LDS copy (16B per lane, tracked by ASYNCcnt) ------
__device__ __forceinline__ void kd_async_copy_b128(float* lds_dst, const float* gsrc) {
  __builtin_amdgcn_global_load_async_to_lds_b128(
      (g_v4i_ptr)(uintptr_t)gsrc,
      (l_v4i_ptr)(uint32_t)(uintptr_t)lds_dst,   // generic LDS addr low 32b == LDS offset
      0, 0);
}

#if defined(__has_builtin) && __has_builtin(__builtin_amdgcn_s_wait_asynccnt)
#define KD_WAIT_ASYNC(N) __builtin_amdgcn_s_wait_asynccnt(N)
#else
#define KD_WAIT_ASYNC(N) asm volatile("s_wait_asynccnt %0" ::"i"(N) : "memory")
#endif

// LDS reads must land in VGPRs before the async engine overwrites the buffer;
// also acts as a compiler memory barrier against reordering.
#define KD_LDS_READS_DONE() asm volatile("s_wait_dscnt 0" ::: "memory")

// ---- online softmax statistics --------------------------------------------
struct KDStat {
  float ms, a1, a2;          // student: max, sum e^(o-ms), sum e^((o-ms)/T)
  float mt, b1, b2, b3, b4;  // teacher: max, sum e^d, sum e^(d/T), sum t*e^d, sum (t-o)*e^(d/T)
};

__device__ __forceinline__ void kd_stat_init(KDStat& s) {
  s.ms = KD_NEGBIG; s.a1 = 0.f; s.a2 = 0.f;
  s.mt = KD_NEGBIG; s.b1 = 0.f; s.b2 = 0.f; s.b3 = 0.f; s.b4 = 0.f;
}

// merge partial o into x (rescale to the larger running max)
__device__ __forceinline__ void kd_stat_merge(KDStat& x, const KDStat& o) {
  {
    float m  = fmaxf(x.ms, o.ms);
    float sx = __expf(x.ms - m),            so  = __expf(o.ms - m);
    float sxT = __expf((x.ms - m) * KD_INVT), soT = __expf((o.ms - m) * KD_INVT);
    x.a1 = x.a1 * sx + o.a1 * so;
    x.a2 = x.a2 * sxT + o.a2 * soT;
    x.ms = m;
  }
  {
    float n  = fmaxf(x.mt, o.mt);
    float tx = __expf(x.mt - n),            to  = __expf(o.mt - n);
    float txT = __expf((x.mt - n) * KD_INVT), toT = __expf((o.mt - n) * KD_INVT);
    x.b1 = x.b1 * tx + o.b1 * to;
    x.b3 = x.b3 * tx + o.b3 * to;
    x.b2 = x.b2 * txT + o.b2 * toT;
    x.b4 = x.b4 * txT + o.b4 * toT;
    x.mt = n;
  }
}

__device__ __forceinline__ KDStat kd_stat_shfl_xor(const KDStat& s, int m) {
  KDStat o;
  o.ms = __shfl_xor(s.ms, m, 32); o.a1 = __shfl_xor(s.a1, m, 32);
  o.a2 = __shfl_xor(s.a2, m, 32); o.mt = __shfl_xor(s.mt, m, 32);
  o.b1 = __shfl_xor(s.b1, m, 32); o.b2 = __shfl_xor(s.b2, m, 32);
  o.b3 = __shfl_xor(s.b3, m, 32); o.b4 = __shfl_xor(s.b4, m, 32);
  return o;
}

// online accumulate one (student, teacher) float4 pair
__device__ __forceinline__ void kd_proc4(KDStat& st, const float4 a, const float4 b) {
  // ---- student ----
  float vm = fmaxf(fmaxf(a.x, a.y), fmaxf(a.z, a.w));
  if (vm > st.ms) {                       // rare: ~log(#chunks) times per thread
    float d = st.ms - vm;
    st.a1 *= __expf(d);
    st.a2 *= __expf(d * KD_INVT);
    st.ms = vm;
  }
  st.a1 += __expf(a.x - st.ms) + __expf(a.y - st.ms) +
           __expf(a.z - st.ms) + __expf(a.w - st.ms);
  st.a2 += __expf((a.x - st.ms) * KD_INVT) + __expf((a.y - st.ms) * KD_INVT) +
           __expf((a.z - st.ms) * KD_INVT) + __expf((a.w - st.ms) * KD_INVT);
  // ---- teacher ----
  float wm = fmaxf(fmaxf(b.x, b.y), fmaxf(b.z, b.w));
  if (wm > st.mt) {
    float d = st.mt - wm;
    float r = __expf(d), rT = __expf(d * KD_INVT);
    st.b1 *= r; st.b3 *= r;
    st.b2 *= rT; st.b4 *= rT;
    st.mt = wm;
  }
  float f1x = __expf(b.x - st.mt), f1y = __expf(b.y - st.mt);
  float f1z = __expf(b.z - st.mt), f1w = __expf(b.w - st.mt);
  st.b1 += f1x + f1y + f1z + f1w;
  st.b3 += b.x * f1x + b.y * f1y + b.z * f1z + b.w * f1w;
  float f2x = __expf((b.x - st.mt) * KD_INVT), f2y = __expf((b.y - st.mt) * KD_INVT);
  float f2z = __expf((b.z - st.mt) * KD_INVT), f2w = __expf((b.w - st.mt) * KD_INVT);
  st.b2 += f2x + f2y + f2z + f2w;
  st.b4 += (b.x - a.x) * f2x + (b.y - a.y) * f2y +
           (b.z - a.z) * f2z + (b.w - a.w) * f2w;
}

// ---- issue one chunk's async copies: exactly 4 async instr/wave, full EXEC -
__device__ __forceinline__ void kd_issue_chunk(const float* gs, const float* gt,
                                               float* bufS, float* bufT,
                                               int c, int tid) {
  const int base = c * KD_CHUNKV4;
  // slot 0: tid in [0,512) always valid
  kd_async_copy_b128(bufS + 4 * tid, gs + 4 * (base + tid));
  kd_async_copy_b128(bufT + 4 * tid, gt + 4 * (base + tid));
  // slot 1: clamp instead of masking -> keeps EXEC all-ones and the per-wave
  // ASYNCcnt accounting exact (duplicate lanes rewrite the same word)
  const int l1 = (tid + KD_BDIM < KD_CHUNKV4) ? (tid + KD_BDIM) : (KD_CHUNKV4 - 1);
  kd_async_copy_b128(bufS + 4 * l1, gs + 4 * (base + l1));
  kd_async_copy_b128(bufT + 4 * l1, gt + 4 * (base + l1));
}

// ---- kernel 1: one block per row, per-row loss -> d_ws ---------------------
extern "C" __global__ void __launch_bounds__(KD_BDIM)
kd_row_kernel(const float* __restrict__ outs,     // [B, C] student logits
              const float* __restrict__ touts,    // [B, C] teacher logits
              const int*   __restrict__ tgts,     // [B] labels
              float*       __restrict__ row_loss) // [B]
{
  // 2 tensors x 2 buffers x 16000 B + scratch = 64.5 KB static LDS
  __shared__ float bufS[2][KD_CHUNKV4 * 4];
  __shared__ float bufT[2][KD_CHUNKV4 * 4];
  __shared__ float scr[16 * 8];

  const int tid = threadIdx.x;
  const int row = blockIdx.x;
  const float* gs = outs  + (size_t)row * KD_C;
  const float* gt = touts + (size_t)row * KD_C;

  KDStat st;
  kd_stat_init(st);

  // prime the pipeline: two chunks in flight (8 async instr/wave, cnt<=63)
  kd_issue_chunk(gs, gt, bufS[0], bufT[0], 0, tid);
  kd_issue_chunk(gs, gt, bufS[1], bufT[1], 1, tid);

  for (int c = 0; c < KD_NCHUNK; ++c) {
    // wait for the oldest chunk only (async loads complete in order per wave;
    // the <=4 remaining ops belong to the newer in-flight chunk)
    if (c < KD_NCHUNK - 1) { KD_WAIT_ASYNC(4); } else { KD_WAIT_ASYNC(0); }

    const float4* s4 = (const float4*)bufS[c & 1];
    const float4* t4 = (const float4*)bufT[c & 1];

    // each lane consumes exactly the words it loaded -> no barrier needed
    kd_proc4(st, s4[tid], t4[tid]);
    if (tid + KD_BDIM < KD_CHUNKV4)
      kd_proc4(st, s4[tid + KD_BDIM], t4[tid + KD_BDIM]);

    if (c + 2 < KD_NCHUNK) {
      KD_LDS_READS_DONE();   // buffer reads landed before async engine rewrites it
      kd_issue_chunk(gs, gt, bufS[c & 1], bufT[c & 1], c + 2, tid);
    }
  }

  // ---- combine per-thread partials: wave32 shfl merge, then cross-wave -----
#pragma unroll
  for (int m = 16; m > 0; m >>= 1) {
    KDStat o = kd_stat_shfl_xor(st, m);
    kd_stat_merge(st, o);
  }
  const int wid = tid >> 5, lane = tid & 31;
  if (lane == 0) {
    float* p = scr + wid * 8;
    p[0] = st.ms; p[1] = st.a1; p[2] = st.a2; p[3] = st.mt;
    p[4] = st.b1; p[5] = st.b2; p[6] = st.b3; p[7] = st.b4;
  }
  __syncthreads();

  if (wid == 0) {
    const float* p = scr + (lane & 15) * 8;   // lanes 16-31 mirror 0-15
    KDStat x;
    x.ms = p[0]; x.a1 = p[1]; x.a2 = p[2]; x.mt = p[3];
    x.b1 = p[4]; x.b2 = p[5]; x.b3 = p[6]; x.b4 = p[7];
#pragma unroll
    for (int m = 8; m > 0; m >>= 1) {        // lanes 0-15 only exchange among themselves
      KDStat o = kd_stat_shfl_xor(x, m);
      kd_stat_merge(x, o);
    }
    if (lane == 0) {
      const float lse_s  = x.ms + logf(x.a1);            // student lse (T=1)
      const float lse_sT = x.ms * KD_INVT + logf(x.a2);  // student lse (temp T)
      const float lse_t  = x.mt + logf(x.b1);            // teacher lse (T=1)
      const float lse_tT = x.mt * KD_INVT + logf(x.b2);  // teacher lse (temp T)

      // alpha = clip(1 - H/log C), H = lse_t - E_p[t]
      const float H = lse_t - x.b3 / x.b1;
      float alpha   = 1.0f - H / logf((float)KD_C);
      alpha         = fminf(1.0f, fmaxf(0.0f, alpha));

      // CE at the label (single L2-hot global load)
      const int   tg = tgts[row];
      const float ce = lse_s - gs[tg];

      // KL(sum) = (1/T) * E_pt[t - o] + lse_sT - lse_tT
      const float kl = (x.b4 / x.b2) * KD_INVT + lse_sT - lse_tT;

      row_loss[row] = (1.0f - alpha) * ce + alpha * (KD_T * KD_T) * kl;
    }
  }
}

// ---- kernel 2: deterministic mean of 4096 row losses -> d_out[0] -----------
extern "C" __global__ void __launch_bounds__(KD_BDIM)
kd_mean_kernel(const float* __restrict__ row_loss, float* __restrict__ out, int n)
{
  __shared__ float red[16];
  const int tid = threadIdx.x;
  float s = 0.f;
  for (int i = tid; i < n; i += KD_BDIM) s += row_loss[i];
#pragma unroll
  for (int m = 16; m > 0; m >>= 1) s += __shfl_xor(s, m, 32);
  const int wid = tid >> 5, lane = tid & 31;
  if (lane == 0) red[wid] = s;
  __syncthreads();
  if (wid == 0) {
    float x = red[lane & 15];                // lanes 16-31 mirror 0-15
#pragma unroll
    for (int m = 8; m > 0; m >>= 1) x += __shfl_xor(x, m, 32);
    if (lane == 0) out[0] = x / (float)n;
  }
}

// ---------------------------------------------------------------------------
extern "C" void kernel_launch(void* const* d_in, const int* in_sizes, int n_in,
                              void* d_out, int out_size, void* d_ws, size_t ws_size,
                              hipStream_t stream) {
  const float* outs  = (const float*)d_in[0];
  const float* touts = (const float*)d_in[1];
  const int*   tgts  = (const int*)d_in[2];   // harness convention: integer -> int*
  const int B = in_sizes[2];                  // 4096 rows

  float* row_loss = (float*)d_ws;             // B floats of scratch

  kd_row_kernel<<<B, KD_BDIM, 0, stream>>>(outs, touts, tgts, row_loss);
  kd_mean_kernel<<<1, KD_BDIM, 0, stream>>>(row_loss, (float*)d_out, B);
}